// ThermalLoss_74414603370949
// MI455X (gfx1250) — compile-verified
//
#include <hip/hip_runtime.h>
#include <math.h>

// ---------- CDNA5 vector types ----------
typedef float v2f __attribute__((ext_vector_type(2)));
typedef float v8f __attribute__((ext_vector_type(8)));

#define IMG_W 1024
#define IMG_H 1024
#define NBATCH 16
#define TILE 64
#define HALO 5
#define SRC_DIM 80          // padded LDS tile rows/cols (74 valid + K-chunk overrun)
#define SP 80               // src LDS row stride (floats)
#define HB_ROWS 80
#define SH 68               // hbuf LDS row stride (floats)
#define NTHREADS 256
#define NWAVES 8
#define NBLOCKS (16*16*16)
#define NPIX (16.0f*1024.0f*1024.0f)

// This toolchain's async-DMA builtins take generic `int*` pointers.
#define IPTR(p) ((int*)(p))

// Normalized 1-D Gaussian, sigma=1.5, 11 taps (matches reference window)
__device__ __constant__ float G11[11] = {
    0.00102838f, 0.00759876f, 0.03600070f, 0.10936060f, 0.21300550f,
    0.26601170f,
    0.21300550f, 0.10936060f, 0.03600070f, 0.00759876f, 0.00102838f
};

__device__ __forceinline__ v8f wmma4(v2f a, v2f b, v8f c) {
    // D = A(16x4 f32) * B(4x16 f32) + C(16x16 f32)
    return __builtin_amdgcn_wmma_f32_16x16x4_f32(
        false, a, false, b, (short)0, c, false, false);
}

// Horizontal Gaussian pass: hbuf[r][c] = sum_k g[k]*src[r][c+k] (src pre-shifted by +HALO)
// 5 row-blocks x 4 col-blocks = 20 WMMA tiles over 80x64
__device__ __forceinline__ void hpass(const float* __restrict__ src,
                                      float* __restrict__ hb,
                                      const v2f* __restrict__ w7,
                                      int wave, int l16, int hsel) {
    for (int t = wave; t < 20; t += NWAVES) {
        const int r0 = (t >> 2) * 16;
        const int c0 = (t & 3) * 16;
        v8f acc = {};
        const int row = r0 + l16;
#pragma unroll
        for (int q = 0; q < 7; ++q) {
            const int col = c0 + q * 4 + hsel * 2;   // even, 8B aligned
            v2f a = *(const v2f*)&src[row * SP + col];
            acc = wmma4(a, w7[q], acc);              // data x banded-weights
        }
        const int sr = r0 + hsel * 8;
        const int sc = c0 + l16;
#pragma unroll
        for (int i = 0; i < 8; ++i)
            hb[(sr + i) * SH + sc] = acc[i];
    }
}

// Vertical Gaussian pass: each wave produces 2 of 16 sub-tiles; accumulates into oA/oB
__device__ __forceinline__ void vpass(const float* __restrict__ hb,
                                      const v2f* __restrict__ w7,
                                      int wave, int l16, int hsel,
                                      v8f& oA, v8f& oB) {
#pragma unroll
    for (int s = 0; s < 2; ++s) {
        const int st = wave + s * 8;
        const int m0 = (st >> 2) * 16;
        const int n0 = (st & 3) * 16;
        v8f acc = s ? oB : oA;
#pragma unroll
        for (int q = 0; q < 7; ++q) {
            const int r = m0 + q * 4 + hsel * 2;
            v2f b;
            b[0] = hb[r * SH + n0 + l16];
            b[1] = hb[(r + 1) * SH + n0 + l16];
            acc = wmma4(w7[q], b, acc);              // banded-weights x data
        }
        if (s) oB = acc; else oA = acc;
    }
}

__global__ __launch_bounds__(NTHREADS)
void thermal_loss_tile_kernel(const float* __restrict__ pred,
                              const float* __restrict__ targ,
                              float* __restrict__ partial) {
    __shared__ float sP[SRC_DIM * SP];
    __shared__ float sT[SRC_DIM * SP];
    __shared__ float sQ[SRC_DIM * SP];
    __shared__ float sH[HB_ROWS * SH];
    __shared__ float sRed[NWAVES * 3];

    const int tid  = threadIdx.x;
    const int wave = tid >> 5;
    const int lane = tid & 31;
    const int l16  = lane & 15;
    const int hsel = (lane >> 4) & 1;

    const int tx0 = blockIdx.x * TILE;
    const int ty0 = blockIdx.y * TILE;
    const long long imgbase = (long long)blockIdx.z * (long long)(IMG_H * IMG_W);
    const float* __restrict__ pim = pred + imgbase;
    const float* __restrict__ tim = targ + imgbase;

    // Interior blocks: whole 80x80 LDS window maps to in-bounds global data.
    const bool interior = (blockIdx.x >= 1) & (blockIdx.x <= 14) &
                          (blockIdx.y >= 1) & (blockIdx.y <= 14);

    if (interior) {
        // ---- CDNA5 async DMA global -> LDS (ASYNCcnt tracked, no VGPR staging) ----
        const float* __restrict__ pbase = &pim[(long long)(ty0 - HALO) * IMG_W + (tx0 - HALO)];
        const float* __restrict__ tbase = &tim[(long long)(ty0 - HALO) * IMG_W + (tx0 - HALO)];
        for (int idx = tid; idx < SRC_DIM * SP; idx += NTHREADS) {
            const int sr = idx / SP, sc = idx % SP;
            const long long o = (long long)sr * IMG_W + sc;
            __builtin_amdgcn_global_load_async_to_lds_b32(IPTR(pbase + o), IPTR(&sP[idx]), 0, 0);
            __builtin_amdgcn_global_load_async_to_lds_b32(IPTR(tbase + o), IPTR(&sT[idx]), 0, 0);
        }
        __builtin_amdgcn_s_wait_asynccnt(0);
    } else {
        // ---- border blocks: checked loads, zero-fill outside image (SAME padding) ----
        for (int idx = tid; idx < SRC_DIM * SP; idx += NTHREADS) {
            const int sr = idx / SP, sc = idx % SP;
            const int gy = ty0 + sr - HALO, gx = tx0 + sc - HALO;
            float pv = 0.f, tv = 0.f;
            if (gy >= 0 && gy < IMG_H && gx >= 0 && gx < IMG_W) {
                const long long o = (long long)gy * IMG_W + gx;
                pv = pim[o];
                tv = tim[o];
            }
            sP[idx] = pv;
            sT[idx] = tv;
        }
    }
    __syncthreads();

    // ---- per-lane banded Gaussian weight chunks (shared by h- and v-pass) ----
    v2f w7[7];
#pragma unroll
    for (int q = 0; q < 7; ++q) {
        const int b0 = q * 4 + hsel * 2;
        const int d0 = b0 - l16, d1 = b0 + 1 - l16;
        float a = 0.f, b = 0.f;
        if (d0 >= 0 && d0 <= 10) a = G11[d0];
        if (d1 >= 0 && d1 <= 10) b = G11[d1];
        w7[q][0] = a;
        w7[q][1] = b;
    }

    // ---- L1 + Sobel edge terms (VALU, from LDS tiles) ----
    float l1s = 0.f, eds = 0.f, sss = 0.f;
#pragma unroll 4
    for (int i = 0; i < 16; ++i) {
        const int pix = tid + NTHREADS * i;      // owned pixel in 64x64 tile
        const int py = pix >> 6, px = pix & 63;
        const float* Pp = &sP[(py + HALO) * SP + (px + HALO)];
        const float* Tp = &sT[(py + HALO) * SP + (px + HALO)];
        l1s += fabsf(Pp[0] - Tp[0]);

        float a0 = Pp[-SP - 1], a1 = Pp[-SP], a2 = Pp[-SP + 1];
        float b0_ = Pp[-1],                   b2 = Pp[1];
        float c0_ = Pp[SP - 1], c1 = Pp[SP],  c2 = Pp[SP + 1];
        float ex = (a2 + 2.f * b2 + c2) - (a0 + 2.f * b0_ + c0_);
        float ey = (c0_ + 2.f * c1 + c2) - (a0 + 2.f * a1 + a2);
        float ep = sqrtf(ex * ex + ey * ey);

        a0 = Tp[-SP - 1]; a1 = Tp[-SP]; a2 = Tp[-SP + 1];
        b0_ = Tp[-1];                    b2 = Tp[1];
        c0_ = Tp[SP - 1]; c1 = Tp[SP];   c2 = Tp[SP + 1];
        ex = (a2 + 2.f * b2 + c2) - (a0 + 2.f * b0_ + c0_);
        ey = (c0_ + 2.f * c1 + c2) - (a0 + 2.f * a1 + a2);
        float et = sqrtf(ex * ex + ey * ey);

        eds += fabsf(ep - et);
    }

    // ---- mu1 = G*pred ----
    v8f mu1a = {}, mu1b = {};
    hpass(sP, sH, w7, wave, l16, hsel);
    __syncthreads();
    vpass(sH, w7, wave, l16, hsel, mu1a, mu1b);
    __syncthreads();

    // ---- mu2 = G*target ----
    v8f mu2a = {}, mu2b = {};
    hpass(sT, sH, w7, wave, l16, hsel);
    __syncthreads();
    vpass(sH, w7, wave, l16, hsel, mu2a, mu2b);
    for (int idx = tid; idx < SRC_DIM * SP; idx += NTHREADS) {
        const float v = sP[idx];
        sQ[idx] = v * v;                 // p^2 (sH not touched)
    }
    __syncthreads();

    // ---- S = G*(p^2) ----
    v8f Sa = {}, Sb = {};
    hpass(sQ, sH, w7, wave, l16, hsel);
    __syncthreads();
    vpass(sH, w7, wave, l16, hsel, Sa, Sb);
    for (int idx = tid; idx < SRC_DIM * SP; idx += NTHREADS) {
        const float v = sT[idx];
        sQ[idx] = v * v;                 // t^2
    }
    __syncthreads();

    // ---- S += G*(t^2)  (accumulate through WMMA C operand) ----
    hpass(sQ, sH, w7, wave, l16, hsel);
    __syncthreads();
    vpass(sH, w7, wave, l16, hsel, Sa, Sb);
    for (int idx = tid; idx < SRC_DIM * SP; idx += NTHREADS)
        sQ[idx] = sP[idx] * sT[idx];     // p*t
    __syncthreads();

    // ---- spt = G*(p*t) ----
    v8f pta = {}, ptb = {};
    hpass(sQ, sH, w7, wave, l16, hsel);
    __syncthreads();
    vpass(sH, w7, wave, l16, hsel, pta, ptb);

    // ---- SSIM map combine (element-wise on WMMA accumulators) ----
    const float C1 = 1e-4f, C2 = 9e-4f, EPS = 1e-6f;
#pragma unroll
    for (int i = 0; i < 8; ++i) {
        {
            const float m1 = mu1a[i], m2 = mu2a[i];
            const float m12 = m1 * m2, m1q = m1 * m1, m2q = m2 * m2;
            const float s12 = pta[i] - m12;
            const float ssum = Sa[i] - m1q - m2q;
            const float num = (2.f * m12 + C1) * (2.f * s12 + C2);
            const float den = (m1q + m2q + C1) * (ssum + C2) + EPS;
            sss += num / den;
        }
        {
            const float m1 = mu1b[i], m2 = mu2b[i];
            const float m12 = m1 * m2, m1q = m1 * m1, m2q = m2 * m2;
            const float s12 = ptb[i] - m12;
            const float ssum = Sb[i] - m1q - m2q;
            const float num = (2.f * m12 + C1) * (2.f * s12 + C2);
            const float den = (m1q + m2q + C1) * (ssum + C2) + EPS;
            sss += num / den;
        }
    }

    // ---- deterministic block reduction: wave shuffle then LDS ----
#pragma unroll
    for (int off = 16; off > 0; off >>= 1) {
        l1s += __shfl_down(l1s, off, 32);
        sss += __shfl_down(sss, off, 32);
        eds += __shfl_down(eds, off, 32);
    }
    if (lane == 0) {
        sRed[wave * 3 + 0] = l1s;
        sRed[wave * 3 + 1] = sss;
        sRed[wave * 3 + 2] = eds;
    }
    __syncthreads();
    if (tid == 0) {
        float a = 0.f, b = 0.f, c = 0.f;
#pragma unroll
        for (int w = 0; w < NWAVES; ++w) {
            a += sRed[w * 3 + 0];
            b += sRed[w * 3 + 1];
            c += sRed[w * 3 + 2];
        }
        const int bid = ((int)blockIdx.z * 16 + (int)blockIdx.y) * 16 + (int)blockIdx.x;
        partial[bid * 3 + 0] = a;   // sum |p-t|
        partial[bid * 3 + 1] = b;   // sum ssim_map
        partial[bid * 3 + 2] = c;   // sum |edge_p - edge_t|
    }
}

__device__ __forceinline__ float guardv(float x) {
    return (__builtin_isnan(x) || __builtin_isinf(x)) ? 0.f : x;
}

__global__ __launch_bounds__(NTHREADS)
void thermal_loss_reduce_kernel(const float* __restrict__ partial,
                                float* __restrict__ out) {
    __shared__ float s[NTHREADS * 3];
    const int tid = threadIdx.x;
    float a = 0.f, b = 0.f, c = 0.f;
    for (int i = tid; i < NBLOCKS; i += NTHREADS) {
        a += partial[i * 3 + 0];
        b += partial[i * 3 + 1];
        c += partial[i * 3 + 2];
    }
    s[tid] = a; s[NTHREADS + tid] = b; s[2 * NTHREADS + tid] = c;
    __syncthreads();
    for (int off = NTHREADS / 2; off > 0; off >>= 1) {
        if (tid < off) {
            s[tid] += s[tid + off];
            s[NTHREADS + tid] += s[NTHREADS + tid + off];
            s[2 * NTHREADS + tid] += s[2 * NTHREADS + tid + off];
        }
        __syncthreads();
    }
    if (tid == 0) {
        const float inv = 1.f / NPIX;
        const float l1   = guardv(s[0] * inv);
        const float ssim = guardv(1.f - s[NTHREADS] * inv);
        const float edge = guardv(s[2 * NTHREADS] * inv);
        const float total = guardv(1.0f * l1 + 0.1f * ssim + 0.1f * edge);
        out[0] = total; out[1] = l1; out[2] = ssim; out[3] = edge;
    }
}

extern "C" void kernel_launch(void* const* d_in, const int* in_sizes, int n_in,
                              void* d_out, int out_size, void* d_ws, size_t ws_size,
                              hipStream_t stream) {
    const float* pred = (const float*)d_in[0];
    const float* targ = (const float*)d_in[1];
    float* out = (float*)d_out;
    float* partial = (float*)d_ws;    // NBLOCKS*3 floats = 48 KB scratch

    dim3 grid(16, 16, 16);            // tiles_x, tiles_y, batch
    thermal_loss_tile_kernel<<<grid, NTHREADS, 0, stream>>>(pred, targ, partial);
    thermal_loss_reduce_kernel<<<1, NTHREADS, 0, stream>>>(partial, out);
}